// CuspModule_38062000177654
// MI455X (gfx1250) — compile-verified
//
#include <hip/hip_runtime.h>
#include <hip/hip_bf16.h>
#include <math.h>

#define N_NODES 50000
#define N_EDGES 800000
#define DIN     128
#define HDIM    128
#define DC      16
#define L_HOPS  3
#define HS      144      // H + DC, row stride for wide buffers

typedef float v2f __attribute__((ext_vector_type(2)));
typedef float v8f __attribute__((ext_vector_type(8)));

// ---------------------------------------------------------------- utilities

__global__ void k_fill(float* __restrict__ p, float v, int n) {
    int i = blockIdx.x * blockDim.x + threadIdx.x;
    if (i < n) p[i] = v;
}

__global__ void k_deg_count(const int* __restrict__ dst, float* __restrict__ deg, int E) {
    int e = blockIdx.x * blockDim.x + threadIdx.x;
    if (e < E) unsafeAtomicAdd(&deg[dst[e]], 1.0f);
}

__global__ void k_edge_w(const int* __restrict__ src, const int* __restrict__ dst,
                         const float* __restrict__ deg, float* __restrict__ wn, int E) {
    int e = blockIdx.x * blockDim.x + threadIdx.x;
    if (e < E) wn[e] = rsqrtf(deg[src[e]] * deg[dst[e]] + 1e-12f);
}

__global__ void k_selfw(const float* __restrict__ deg, float* __restrict__ sw, int n) {
    int i = blockIdx.x * blockDim.x + threadIdx.x;
    if (i < n) sw[i] = rsqrtf(deg[i] * deg[i] + 1e-12f);
}

// ce row: node_orc == 0 -> ce = relu(ce_b1) @ ce_W2 + ce_b2  (same for all nodes)
__global__ void k_ce_row(const float* __restrict__ ceb1,   // [64]
                         const float* __restrict__ ceW2,   // [64 x 16]
                         const float* __restrict__ ceb2,   // [16]
                         float* __restrict__ cer) {        // [16]
    int o = threadIdx.x;
    if (o < DC) {
        float s = ceb2[o];
        for (int j = 0; j < 64; ++j)
            s += fmaxf(ceb1[j], 0.0f) * ceW2[j * DC + o];
        cer[o] = s;
    }
}

__global__ void k_concat_ce(float* __restrict__ h, const float* __restrict__ cer, int n) {
    int t = blockIdx.x * blockDim.x + threadIdx.x;
    if (t >= n * DC) return;
    int r = t / DC, c = t - r * DC;
    h[(size_t)r * HS + HDIM + c] = cer[c];
}

// ----------------------------------------------------- GPR filter primitives

__global__ void k_scale_rows(float* __restrict__ dst, const float* __restrict__ src,
                             const float* __restrict__ gamma, int gi,
                             int stride, int dim, int n) {
    int t = blockIdx.x * blockDim.x + threadIdx.x;
    if (t >= n * dim) return;
    int r = t / dim, c = t - r * dim;
    dst[(size_t)r * stride + c] = gamma[gi] * src[(size_t)r * stride + c];
}

__global__ void k_axpy_rows(float* __restrict__ dst, const float* __restrict__ src,
                            const float* __restrict__ gamma, int gi,
                            int stride, int dim, int n) {
    int t = blockIdx.x * blockDim.x + threadIdx.x;
    if (t >= n * dim) return;
    int r = t / dim, c = t - r * dim;
    dst[(size_t)r * stride + c] += gamma[gi] * src[(size_t)r * stride + c];
}

__global__ void k_prop_init(float* __restrict__ dst, const float* __restrict__ src,
                            const float* __restrict__ sw,
                            int stride, int dim, int n) {
    int t = blockIdx.x * blockDim.x + threadIdx.x;
    if (t >= n * dim) return;
    int r = t / dim, c = t - r * dim;
    dst[(size_t)r * stride + c] = sw[r] * src[(size_t)r * stride + c];
}

// scatter: out[dst] += w_norm[e] * in[src], 4 features per thread
__global__ void k_prop_edges(const int* __restrict__ src, const int* __restrict__ dst,
                             const float* __restrict__ wn,
                             const float* __restrict__ hin, float* __restrict__ hout,
                             int stride, int dim, int E) {
    int t = blockIdx.x * blockDim.x + threadIdx.x;
    int groups = dim >> 2;
    int e = t / groups;
    if (e >= E) return;
    int g = (t - e * groups) << 2;
    int s = src[e], d = dst[e];
    float w = wn[e];
    const float4 hv = *(const float4*)(hin + (size_t)s * stride + g);
    float* o = hout + (size_t)d * stride + g;
    unsafeAtomicAdd(o + 0, w * hv.x);
    unsafeAtomicAdd(o + 1, w * hv.y);
    unsafeAtomicAdd(o + 2, w * hv.z);
    unsafeAtomicAdd(o + 3, w * hv.w);
}

// ------------------------------------------------------------- WMMA fp32 GEMM
// C[M x Nout] = A[M x K] (row stride SA) @ W[K x Nout] + bias, optional ReLU.
// One block = 16 rows; one wave = one 16x16 output tile (Nout/16 waves/block).
// A tile staged in LDS (padded stride), W streamed from L2/WGP$.
__global__ void k_gemm_wmma(const float* __restrict__ A, int SA,
                            const float* __restrict__ W, const float* __restrict__ bias,
                            float* __restrict__ C, int SC,
                            int M, int K, int Nout, int relu) {
    extern __shared__ float sA[];              // 16 x (K+1)
    const int lane = threadIdx.x & 31;
    const int wave = threadIdx.x >> 5;
    const int row0 = blockIdx.x * 16;
    if (row0 >= M) return;
    const int KP = K + 1;

    for (int idx = threadIdx.x; idx < 16 * K; idx += blockDim.x) {
        int r = idx / K, c = idx - r * K;
        sA[r * KP + c] = A[(size_t)(row0 + r) * SA + c];
    }
    __syncthreads();

    const int half = lane >> 4;                // 0: lanes 0-15, 1: lanes 16-31
    const int l    = lane & 15;
    const int col0 = wave * 16;

    v8f acc = {0.f, 0.f, 0.f, 0.f, 0.f, 0.f, 0.f, 0.f};
    for (int k = 0; k < K; k += 4) {
        const int ka = k + 2 * half;
        v2f a, b;
        // A 16x4 fp32 layout (ISA 7.12.2): lanes 0-15 hold K=k,k+1; lanes 16-31 hold K=k+2,k+3
        a.x = sA[l * KP + ka];
        a.y = sA[l * KP + ka + 1];
        // B 4x16 fp32: N striped across lanes; lane half selects K pair
        b.x = W[(size_t)ka * Nout + col0 + l];
        b.y = W[(size_t)(ka + 1) * Nout + col0 + l];
        acc = __builtin_amdgcn_wmma_f32_16x16x4_f32(
            /*neg_a=*/false, a, /*neg_b=*/false, b,
            /*c_mod=*/(short)0, acc, /*reuse_a=*/false, /*reuse_b=*/false);
    }

    const float bv = bias ? bias[col0 + l] : 0.0f;
    #pragma unroll
    for (int r = 0; r < 8; ++r) {
        int row = row0 + r + 8 * half;         // C/D: VGPR r -> M=r (lo half) / M=r+8 (hi half)
        float v = acc[r] + bv;
        if (relu) v = fmaxf(v, 0.0f);
        C[(size_t)row * SC + col0 + l] = v;
    }
}

// ------------------------------------------------ manifold mix + LN (per node)

__device__ inline float block_red_sum(float v, float* red) {
    int t = threadIdx.x;
    red[t] = v;
    __syncthreads();
    for (int s = 64; s > 0; s >>= 1) {
        if (t < s) red[t] += red[t + s];
        __syncthreads();
    }
    float r = red[0];
    __syncthreads();
    return r;
}

__device__ inline float softplusf(float x) {
    return (x > 20.0f) ? x : log1pf(expf(x));
}

__global__ void k_manifold_fuse(const float* __restrict__ ze, const float* __restrict__ zh,
                                const float* __restrict__ zs, const float* __restrict__ hin,
                                float* __restrict__ hout, int strideIO,
                                const float* __restrict__ hyp_c, const float* __restrict__ sph_c,
                                const float* __restrict__ mixw,
                                const float* __restrict__ ln_s, const float* __restrict__ ln_b,
                                int residual) {
    __shared__ float red[128];
    const int i = blockIdx.x;
    const int f = threadIdx.x;
    const size_t b = (size_t)i * HDIM;
    float e  = ze[b + f];
    float hv = zh[b + f];
    float sv = zs[b + f];

    float ssh = block_red_sum(hv * hv, red);
    float sss = block_red_sum(sv * sv, red);

    float sqh = sqrtf(fmaxf(softplusf(hyp_c[0]), 1e-12f));
    float sqs = sqrtf(fmaxf(softplusf(sph_c[0]), 1e-12f));
    float ah = sqh * fmaxf(sqrtf(ssh), 1e-6f);
    float as = sqs * fmaxf(sqrtf(sss), 1e-6f);
    float fh = tanhf(ah) / ah;
    float fs = sinf(as) / as;

    float m0 = mixw[0], m1 = mixw[1], m2 = mixw[2];
    float mx = fmaxf(m0, fmaxf(m1, m2));
    float e0 = expf(m0 - mx), e1 = expf(m1 - mx), e2 = expf(m2 - mx);
    float inv = 1.0f / (e0 + e1 + e2);

    float val = inv * (e0 * e + e1 * fh * hv + e2 * fs * sv);
    if (residual) val += hin[(size_t)i * strideIO + f];

    float mu  = block_red_sum(val, red) * (1.0f / HDIM);
    float d   = val - mu;
    float var = block_red_sum(d * d, red) * (1.0f / HDIM);
    hout[(size_t)i * strideIO + f] = d * rsqrtf(var + 1e-5f) * ln_s[f] + ln_b[f];
}

// --------------------------------------------------------------------- launch

extern "C" void kernel_launch(void* const* d_in, const int* in_sizes, int n_in,
                              void* d_out, int out_size, void* d_ws, size_t ws_size,
                              hipStream_t stream) {
    // jax pytree (sorted-key) flatten order of setup_inputs():
    const float* x      = (const float*)d_in[0];
    const int*   ei     = (const int*)d_in[1];
    const float* W_in   = (const float*)d_in[2];
    const float* b_in   = (const float*)d_in[3];
    // ce: ce_W1(idx4, unused since orc==0), ce_W2, ce_b1, ce_b2
    const float* ce_W2  = (const float*)d_in[5];
    const float* ce_b1  = (const float*)d_in[6];
    const float* ce_b2  = (const float*)d_in[7];
    // per-layer (12 leaves): eW,eb,gamma,hW,hb,hc,ln_b,ln_s,mixw,sW,sb,sc
    const int LBASE = 8, LSTRIDE = 12;
    const float* out_W1 = (const float*)d_in[LBASE + 2 * LSTRIDE + 0];
    const float* out_W2 = (const float*)d_in[LBASE + 2 * LSTRIDE + 1];
    const float* out_b1 = (const float*)d_in[LBASE + 2 * LSTRIDE + 2];
    const float* out_b2 = (const float*)d_in[LBASE + 2 * LSTRIDE + 3];

    const int* src = ei;
    const int* dst = ei + N_EDGES;

    // workspace layout (floats, 64-float aligned)
    float* ws   = (float*)d_ws;
    float* deg  = ws;                       // 50048
    float* sw   = deg  + 50048;             // 50048
    float* wn   = sw   + 50048;             // 800000
    float* cer  = wn   + 800000;            // 64
    float* h    = cer  + 64;                // N*HS
    float* acc  = h    + (size_t)N_NODES * HS;
    float* bufA = acc  + (size_t)N_NODES * HS;
    float* bufB = bufA + (size_t)N_NODES * HS;
    float* zs   = bufB + (size_t)N_NODES * HS;  // N*HDIM

    const int T = 256;
    auto blocks = [](long long n, int t) { return (unsigned)((n + t - 1) / t); };

    // ---- graph weights
    k_fill<<<blocks(N_NODES, T), T, 0, stream>>>(deg, 1.0f, N_NODES);
    k_deg_count<<<blocks(N_EDGES, T), T, 0, stream>>>(dst, deg, N_EDGES);
    k_selfw<<<blocks(N_NODES, T), T, 0, stream>>>(deg, sw, N_NODES);
    k_edge_w<<<blocks(N_EDGES, T), T, 0, stream>>>(src, dst, deg, wn, N_EDGES);

    // ---- input projection + constant curvature encoding
    auto gemm = [&](const float* A, int SA, const float* W, const float* bias,
                    float* C, int SC, int M, int K, int Nout, int relu) {
        dim3 grid((M + 15) / 16);
        dim3 block(32 * (Nout / 16));
        size_t lds = (size_t)16 * (K + 1) * sizeof(float);
        k_gemm_wmma<<<grid, block, lds, stream>>>(A, SA, W, bias, C, SC, M, K, Nout, relu);
    };

    gemm(x, DIN, W_in, b_in, h, HS, N_NODES, DIN, HDIM, 0);
    k_ce_row<<<1, 32, 0, stream>>>(ce_b1, ce_W2, ce_b2, cer);
    k_concat_ce<<<blocks((long long)N_NODES * DC, T), T, 0, stream>>>(h, cer, N_NODES);

    // ---- layers
    for (int li = 0; li < 2; ++li) {
        const float* const* lp = (const float* const*)&d_in[LBASE + li * LSTRIDE];
        const float* eW = lp[0];  const float* eb = lp[1];
        const float* gm = lp[2];
        const float* hW = lp[3];  const float* hb = lp[4];  const float* hc = lp[5];
        const float* lnb = lp[6]; const float* lns = lp[7];
        const float* mw = lp[8];
        const float* sW = lp[9];  const float* sb = lp[10]; const float* sc = lp[11];

        const int din = (li == 0) ? (HDIM + DC) : HDIM;
        const long long nd = (long long)N_NODES * din;

        // GPR filter: acc = sum_k gamma_k P^k h
        k_scale_rows<<<blocks(nd, T), T, 0, stream>>>(acc, h, gm, 0, HS, din, N_NODES);
        const float* cur = h;
        for (int k = 1; k <= L_HOPS; ++k) {
            float* pk = (k == 2) ? bufB : bufA;
            k_prop_init<<<blocks(nd, T), T, 0, stream>>>(pk, cur, sw, HS, din, N_NODES);
            k_prop_edges<<<blocks((long long)N_EDGES * (din >> 2), T), T, 0, stream>>>(
                src, dst, wn, cur, pk, HS, din, N_EDGES);
            k_axpy_rows<<<blocks(nd, T), T, 0, stream>>>(acc, pk, gm, k, HS, din, N_NODES);
            cur = pk;
        }

        // product-manifold linears (WMMA), then fused mix + residual + LayerNorm
        gemm(acc, HS, eW, eb, bufA, HDIM, N_NODES, din, HDIM, 0);  // euclidean
        gemm(acc, HS, hW, hb, bufB, HDIM, N_NODES, din, HDIM, 0);  // hyperbolic
        gemm(acc, HS, sW, sb, zs,   HDIM, N_NODES, din, HDIM, 0);  // spherical
        k_manifold_fuse<<<N_NODES, 128, 0, stream>>>(
            bufA, bufB, zs, h, h, HS, hc, sc, mw, lns, lnb, /*residual=*/(li > 0) ? 1 : 0);
    }

    // ---- output head
    gemm(h, HS, out_W1, out_b1, bufB, 64, N_NODES, HDIM, 64, 1);   // relu
    gemm(bufB, 64, out_W2, out_b2, (float*)d_out, 64, N_NODES, 64, 64, 0);
}